// ImageGraphConvolution_14250701488407
// MI455X (gfx1250) — compile-verified
//
#include <hip/hip_runtime.h>

typedef __attribute__((ext_vector_type(2))) float v2f;
typedef __attribute__((ext_vector_type(8))) float v8f;

#define N_NODES 100000
#define N_EDGES 3200000
#define FEAT    1024
#define OUT_DIM 14
#define PAD_DIM 16   // support rows padded to 16 floats (64 B) in workspace

// ---------------------------------------------------------------------------
// Kernel 1: support = x @ W + b  via V_WMMA_F32_16X16X4_F32 (wave32).
// Block = 256 threads = 8 waves; wave w computes rows [base, base+16) x 16 cols.
// HBM-bound on streaming x (410 MB -> ~17.6 us floor @ 23.3 TB/s).
// ---------------------------------------------------------------------------
__global__ __launch_bounds__(256)
void gemm_support_kernel(const float* __restrict__ x,
                         const float* __restrict__ W,
                         const float* __restrict__ bias,
                         float* __restrict__ support) {
    // W staged in LDS, natural 14-wide layout: 1024*14*4 = 56 KB.
    __shared__ float Wlds[FEAT * OUT_DIM];

    const int tid = threadIdx.x;
    for (int i = tid; i < FEAT * OUT_DIM; i += 256) {
        Wlds[i] = W[i];
    }
    __syncthreads();   // single barrier; inactive tail waves may return after it

    const int wave = tid >> 5;   // wave32
    const int lane = tid & 31;
    const int tile = (blockIdx.x * 8 + wave) * 16;   // first row of this tile
    if (tile >= N_NODES) return;                      // wave-uniform

    // ISA layout for 32-bit 16x4 A (and mirrored 4x16 B):
    //   lanes 0-15: M/N = lane, VGPR0 = K+0, VGPR1 = K+1
    //   lanes 16-31: M/N = lane-16, VGPR0 = K+2, VGPR1 = K+3
    const int m  = lane & 15;          // row of A / col of B,C
    const int kb = (lane >> 4) * 2;    // K sub-offset for this half-wave

    const float* xrow = x + (size_t)(tile + m) * FEAT + kb;

    const int widx = (m < OUT_DIM) ? m : 0;   // clamp; zero-selected below

    v8f c = {};
    #pragma unroll 4
    for (int k0 = 0; k0 < FEAT; k0 += 4) {
        // A fragment: two consecutive fp32 -> one global_load_b64 per lane.
        v2f a = *(const v2f*)(xrow + k0);
        // B fragment from LDS (lane banks disjoint mod 64 -> conflict-free).
        v2f b;
        float b0 = Wlds[(k0 + kb)     * OUT_DIM + widx];
        float b1 = Wlds[(k0 + kb + 1) * OUT_DIM + widx];
        b.x = (m < OUT_DIM) ? b0 : 0.0f;
        b.y = (m < OUT_DIM) ? b1 : 0.0f;
        // D = A(16x4) * B(4x16) + C  -> v_wmma_f32_16x16x4_f32
        c = __builtin_amdgcn_wmma_f32_16x16x4_f32(false, a, false, b,
                                                  (short)0, c, false, false);
    }

    const float bv = (m < OUT_DIM) ? bias[m] : 0.0f;
    // C/D layout: VGPR i -> M = i (lanes 0-15) or 8+i (lanes 16-31); N = lane&15.
    const int row_base = tile + ((lane >> 4) << 3);
    #pragma unroll
    for (int i = 0; i < 8; ++i) {
        support[(size_t)(row_base + i) * PAD_DIM + m] = c[i] + bv;
    }
}

// ---------------------------------------------------------------------------
// Kernel 2: zero the output accumulator (harness poisons d_out with 0xAA).
// ---------------------------------------------------------------------------
__global__ __launch_bounds__(256)
void zero_out_kernel(float* __restrict__ out, int n) {
    int i = blockIdx.x * 256 + threadIdx.x;
    if (i < n) out[i] = 0.0f;
}

// ---------------------------------------------------------------------------
// Kernel 3: COO SpMM scatter. 16 lanes per edge: lane j gathers one float of
// support[col] (64 B row -> one coalesced L2 request) and atomically adds
// vals[e]*support[col][j] into out[row][j]. support (6.4 MB) and out (5.6 MB)
// are L2-resident (192 MB L2), so this phase is L2-bound, not HBM-bound.
// ---------------------------------------------------------------------------
__global__ __launch_bounds__(256)
void spmm_scatter_kernel(const float* __restrict__ vals,
                         const int*   __restrict__ rows,
                         const int*   __restrict__ cols,
                         const float* __restrict__ support,
                         float*       __restrict__ out) {
    long long t = (long long)blockIdx.x * 256 + threadIdx.x;
    int e = (int)(t >> 4);
    int j = (int)(t & 15);
    if (e >= N_EDGES || j >= OUT_DIM) return;

    int   r = rows[e];
    int   cidx = cols[e];
    float v = vals[e];
    float s = support[(size_t)cidx * PAD_DIM + j];
    atomicAdd(out + (size_t)r * OUT_DIM + j, v * s);
}

// ---------------------------------------------------------------------------
// Launch: inputs in setup_inputs() order:
//   0: x [100000*1024] f32   1: W_cls [1024*14] f32   2: b_cls [14] f32
//   3: adj_vals [3.2M] f32   4: adj_row [3.2M] i32    5: adj_col [3.2M] i32
// d_out: [100000*14] f32.  d_ws: support padded to 16 cols (6.4 MB).
// ---------------------------------------------------------------------------
extern "C" void kernel_launch(void* const* d_in, const int* in_sizes, int n_in,
                              void* d_out, int out_size, void* d_ws, size_t ws_size,
                              hipStream_t stream) {
    const float* x    = (const float*)d_in[0];
    const float* W    = (const float*)d_in[1];
    const float* b    = (const float*)d_in[2];
    const float* vals = (const float*)d_in[3];
    const int*   rows = (const int*)d_in[4];
    const int*   cols = (const int*)d_in[5];
    float* out        = (float*)d_out;
    float* support    = (float*)d_ws;   // N_NODES * PAD_DIM floats = 6.4 MB

    // 1) dense projection via WMMA
    {
        int rows_per_block = 8 * 16;  // 8 waves x 16-row tiles
        int grid = (N_NODES + rows_per_block - 1) / rows_per_block;  // 782
        gemm_support_kernel<<<grid, 256, 0, stream>>>(x, W, b, support);
    }

    // 2) zero output
    {
        int n = N_NODES * OUT_DIM;
        int grid = (n + 255) / 256;
        zero_out_kernel<<<grid, 256, 0, stream>>>(out, n);
    }

    // 3) edge scatter (atomics into L2-resident output)
    {
        long long threads = (long long)N_EDGES * 16;
        int grid = (int)((threads + 255) / 256);   // 200000 blocks
        spmm_scatter_kernel<<<grid, 256, 0, stream>>>(vals, rows, cols, support, out);
    }
}